// RGCN_48043504173159
// MI455X (gfx1250) — compile-verified
//
#include <hip/hip_runtime.h>

typedef float v2f __attribute__((ext_vector_type(2)));
typedef float v8f __attribute__((ext_vector_type(8)));

#define IN_DIM 128
#define RR 8

// ---------------- zero int buffer ----------------
__global__ void rgcn_zero_i32(int* __restrict__ p, int n) {
    int i = blockIdx.x * blockDim.x + threadIdx.x;
    if (i < n) p[i] = 0;
}

// ---------------- per-(tgt,relation) segment counts ----------------
__global__ void rgcn_count_kernel(const int* __restrict__ tgt,
                                  const int* __restrict__ ety,
                                  int* __restrict__ cntI, int E) {
    int e = blockIdx.x * blockDim.x + threadIdx.x;
    if (e >= E) return;
    atomicAdd(&cntI[tgt[e] * RR + ety[e]], 1);
}

// ---------------- single-block hierarchical exclusive scan over nseg counts ----------------
__global__ void rgcn_scan_kernel(const int* __restrict__ cntI,
                                 int* __restrict__ offsets, int nseg) {
    __shared__ int sh[1024];
    int t = threadIdx.x;
    int chunk = (nseg + 1023) >> 10;
    int lo = t * chunk;
    int hi = lo + chunk; if (hi > nseg) hi = nseg;
    int sum = 0;
    for (int i = lo; i < hi; ++i) sum += cntI[i];
    sh[t] = sum;
    __syncthreads();
    for (int off = 1; off < 1024; off <<= 1) {   // Hillis-Steele inclusive
        int v = (t >= off) ? sh[t - off] : 0;
        __syncthreads();
        sh[t] += v;
        __syncthreads();
    }
    int run = sh[t] - sum;                       // exclusive base for this thread
    for (int i = lo; i < hi; ++i) { offsets[i] = run; run += cntI[i]; }
    if (t == 1023) offsets[nseg] = sh[1023];     // total == E
}

// ---------------- bucket placement: sortedSrc grouped by segment ----------------
__global__ void rgcn_place_kernel(const int* __restrict__ src,
                                  const int* __restrict__ tgt,
                                  const int* __restrict__ ety,
                                  int* __restrict__ cursor,
                                  int* __restrict__ sortedSrc, int E) {
    int e = blockIdx.x * blockDim.x + threadIdx.x;
    if (e >= E) return;
    int seg = tgt[e] * RR + ety[e];
    int pos = atomicAdd(&cursor[seg], 1);
    sortedSrc[pos] = src[e];
}

// ---------------- mean aggregation: one wave per (node,relation) segment ----------------
__global__ void rgcn_aggregate_kernel(const float* __restrict__ x,      // [N,128]
                                      const int* __restrict__ sortedSrc,
                                      const int* __restrict__ offsets,  // [nseg+1]
                                      float* __restrict__ agg,          // [nseg,128]
                                      int nseg) {
    int wave = blockIdx.x * (blockDim.x >> 5) + (threadIdx.x >> 5);
    if (wave >= nseg) return;
    int lane = threadIdx.x & 31;
    int start = offsets[wave];
    int end   = offsets[wave + 1];
    float4 acc = make_float4(0.f, 0.f, 0.f, 0.f);
    for (int j = start; j < end; ++j) {
        int s = sortedSrc[j];                    // wave-uniform broadcast load
        float4 v = *(const float4*)(x + ((long)s << 7) + (lane << 2));
        acc.x += v.x; acc.y += v.y; acc.z += v.z; acc.w += v.w;
    }
    float inv = 1.0f / fmaxf((float)(end - start), 1.0f);
    acc.x *= inv; acc.y *= inv; acc.z *= inv; acc.w *= inv;
    *(float4*)(agg + ((long)wave << 7) + (lane << 2)) = acc;
}

// ---------------- pack B rows into K-pairs so each fragment is one b64 load ----------------
__global__ void rgcn_pack_kernel(const float* __restrict__ W,  // [2*K2, LD]
                                 float2* __restrict__ Wp,      // [K2, LD]
                                 int K2, int LD) {
    int i = blockIdx.x * blockDim.x + threadIdx.x;
    if (i >= K2 * LD) return;
    int k2 = i / LD, c = i - k2 * LD;
    Wp[i] = make_float2(W[(2 * k2) * LD + c], W[(2 * k2 + 1) * LD + c]);
}

// ---------------- WMMA GEMM: out[16 rows x NT*16 cols] per wave ----------------
// out = relu?( [agg | xroot] @ [Wstk ; Wroot] + bias ), K = 1024 + 128
template <int NT>
__launch_bounds__(256)
__global__ void rgcn_gemm_kernel(const float* __restrict__ agg,    // [N, 1024]
                                 const float* __restrict__ xroot,  // [N, 128]
                                 const v2f* __restrict__ Wp,       // [512, NT*16] K-pairs
                                 const v2f* __restrict__ Rp,       // [64,  NT*16] K-pairs
                                 const float* __restrict__ bias,   // [NT*16]
                                 float* __restrict__ outp,         // [N, NT*16]
                                 int ntiles, int relu) {
    const int LD = NT * 16;
    int wave = blockIdx.x * (blockDim.x >> 5) + (threadIdx.x >> 5);
    if (wave >= ntiles) return;  // wave-uniform: EXEC stays all-1s for WMMA
    int lane = threadIdx.x & 31;
    int row0 = wave << 4;
    int m    = lane & 15;        // A-frag row / B-frag col
    int half = lane >> 4;        // 0 or 1 -> K-pair select within 4-wide step
    int koff = half << 1;

    v8f acc[NT] = {};

    // --- region 1: aggregated means vs stacked relation weights (K = 1024) ---
    const float* arow = agg + ((long)(row0 + m)) * (RR * IN_DIM);
#pragma unroll 1
    for (int k0 = 0; k0 < RR * IN_DIM; k0 += 4) {
        v2f a = *(const v2f*)(arow + k0 + koff);
        const v2f* bp = Wp + (long)((k0 >> 1) + half) * LD + m;
#pragma unroll
        for (int t = 0; t < NT; ++t) {
            v2f b = bp[t * 16];
            acc[t] = __builtin_amdgcn_wmma_f32_16x16x4_f32(
                false, a, false, b, (short)0, acc[t], false, false);
        }
    }
    // --- region 2: root transform (K = 128) ---
    const float* xr = xroot + ((long)(row0 + m)) * IN_DIM;
#pragma unroll 1
    for (int k0 = 0; k0 < IN_DIM; k0 += 4) {
        v2f a = *(const v2f*)(xr + k0 + koff);
        const v2f* bp = Rp + (long)((k0 >> 1) + half) * LD + m;
#pragma unroll
        for (int t = 0; t < NT; ++t) {
            v2f b = bp[t * 16];
            acc[t] = __builtin_amdgcn_wmma_f32_16x16x4_f32(
                false, a, false, b, (short)0, acc[t], false, false);
        }
    }
    // --- epilogue: D layout = VGPR v holds M=v (lanes 0-15) / M=v+8 (lanes 16-31) ---
    int mbase = half << 3;
#pragma unroll
    for (int t = 0; t < NT; ++t) {
        int col = t * 16 + m;
        float bv = bias[col];
#pragma unroll
        for (int v = 0; v < 8; ++v) {
            float val = acc[t][v] + bv;
            if (relu) val = fmaxf(val, 0.0f);
            outp[(long)(row0 + mbase + v) * LD + col] = val;
        }
    }
}

extern "C" void kernel_launch(void* const* d_in, const int* in_sizes, int n_in,
                              void* d_out, int out_size, void* d_ws, size_t ws_size,
                              hipStream_t stream) {
    const float* x     = (const float*)d_in[0];
    const float* W1    = (const float*)d_in[1];  // [8,128,128] == [1024,128]
    const float* root1 = (const float*)d_in[2];  // [128,128]
    const float* b1    = (const float*)d_in[3];
    const float* W2    = (const float*)d_in[4];  // [8,128,64] == [1024,64]
    const float* root2 = (const float*)d_in[5];  // [128,64]
    const float* b2    = (const float*)d_in[6];
    const int*   eidx  = (const int*)d_in[7];    // [2,E]
    const int*   etype = (const int*)d_in[8];    // [E]

    int N = in_sizes[0] / IN_DIM;
    int E = in_sizes[8];
    const int* src = eidx;
    const int* tgt = eidx + E;
    int nseg = N * RR;

    // ---- workspace layout ----
    float* agg = (float*)d_ws;                        // N*1024 f32
    float* h1  = agg + (long)N * RR * IN_DIM;         // N*128 f32
    v2f*   W1p = (v2f*)(h1 + (long)N * IN_DIM);       // 512*128 pairs
    v2f*   R1p = W1p + 512 * 128;                     // 64*128 pairs
    v2f*   W2p = R1p + 64 * 128;                      // 512*64 pairs
    v2f*   R2p = W2p + 512 * 64;                      // 64*64 pairs
    int*   offsets   = (int*)(R2p + 64 * 64);         // nseg+1
    int    segAlloc  = (nseg + 8) & ~7;
    int*   cursor    = offsets + segAlloc;            // nseg
    int*   cntI      = cursor + segAlloc;             // nseg
    int*   sortedSrc = cntI + segAlloc;               // E
    float* out = (float*)d_out;

    int eBlocks   = (E + 255) / 256;
    int ntiles    = N / 16;                           // 50000/16 = 3125 exact
    int gemmBlk   = (ntiles + 7) / 8;                 // 8 waves / 256-thread block
    int aggBlocks = (nseg + 7) / 8;                   // 8 waves per block

    // ---- build CSR-by-(tgt,relation) once; reused by both layers ----
    rgcn_zero_i32<<<(nseg + 255) / 256, 256, 0, stream>>>(cntI, nseg);
    rgcn_count_kernel<<<eBlocks, 256, 0, stream>>>(tgt, etype, cntI, E);
    rgcn_scan_kernel<<<1, 1024, 0, stream>>>(cntI, offsets, nseg);
    hipMemcpyAsync(cursor, offsets, (size_t)nseg * sizeof(int),
                   hipMemcpyDeviceToDevice, stream);
    rgcn_place_kernel<<<eBlocks, 256, 0, stream>>>(src, tgt, etype, cursor,
                                                   sortedSrc, E);

    // ---- pack weights into K-pair fragments ----
    rgcn_pack_kernel<<<(512 * 128 + 255) / 256, 256, 0, stream>>>(W1, (float2*)W1p, 512, 128);
    rgcn_pack_kernel<<<(64 * 128 + 255) / 256, 256, 0, stream>>>(root1, (float2*)R1p, 64, 128);
    rgcn_pack_kernel<<<(512 * 64 + 255) / 256, 256, 0, stream>>>(W2, (float2*)W2p, 512, 64);
    rgcn_pack_kernel<<<(64 * 64 + 255) / 256, 256, 0, stream>>>(root2, (float2*)R2p, 64, 64);

    // ---- layer 1 ----
    rgcn_aggregate_kernel<<<aggBlocks, 256, 0, stream>>>(x, sortedSrc, offsets, agg, nseg);
    rgcn_gemm_kernel<8><<<gemmBlk, 256, 0, stream>>>(agg, x, W1p, R1p, b1, h1,
                                                     ntiles, /*relu=*/1);
    // ---- layer 2 ----
    rgcn_aggregate_kernel<<<aggBlocks, 256, 0, stream>>>(h1, sortedSrc, offsets, agg, nseg);
    rgcn_gemm_kernel<4><<<gemmBlk, 256, 0, stream>>>(agg, h1, W2p, R2p, b2, out,
                                                     ntiles, /*relu=*/0);
}